// CircuitEmbedding_83210696393026
// MI455X (gfx1250) — compile-verified
//
#include <hip/hip_runtime.h>
#include <hip/hip_bf16.h>
#include <math.h>

#define H 128
#define EPS 1e-5f

typedef __attribute__((ext_vector_type(2))) float v2f;
typedef __attribute__((ext_vector_type(8))) float v8f;
typedef __attribute__((ext_vector_type(4))) unsigned int v4u;
typedef __attribute__((ext_vector_type(8))) int v8i;
typedef __attribute__((ext_vector_type(4))) int v4i;

// ---------------------------------------------------------------- utilities
__global__ void k_fill(float* __restrict__ p, float v, int n) {
    int i = blockIdx.x * blockDim.x + threadIdx.x;
    if (i < n) p[i] = v;
}

__device__ inline void atomicMaxF(float* addr, float v) {
    int* ia = (int*)addr;
    int old = __float_as_int(*addr);
    while (__int_as_float(old) < v) {
        int assumed = old;
        old = atomicCAS(ia, assumed, __float_as_int(v));
        if (old == assumed) break;
    }
}

// ---------------------------------------------------------------- degrees
__global__ void k_deg_count(const int* __restrict__ dst, float* __restrict__ deg, int E) {
    int e = blockIdx.x * blockDim.x + threadIdx.x;
    if (e < E) atomicAdd(&deg[dst[e]], 1.0f);
}

__global__ void k_dinv(float* __restrict__ deg, int N) {
    int i = blockIdx.x * blockDim.x + threadIdx.x;
    if (i < N) deg[i] = rsqrtf(deg[i]);
}

// ---------------------------------------------------------------- WMMA GEMM
// out[M x 128] = act(A[M x 128] @ W[128 x 128] + bias)
// act: 0=none, 1=relu, 2=tanh. bias may be nullptr.
// Block = 256 threads = 8 waves; each wave computes one 16x16 tile via
// v_wmma_f32_16x16x4_f32. A-tile is DMA'd into LDS by the Tensor Data Mover
// (with TDM padding producing the bank-conflict-free 132-float row pitch);
// W is staged transposed in LDS so both WMMA operands come from ds_load_b64.
__global__ __launch_bounds__(256)
void k_gemm128(const float* __restrict__ A, const float* __restrict__ W,
               const float* __restrict__ bias, float* __restrict__ out,
               int M, int act) {
    // 132-float pitch: bank = (4*row + k) % 64 -> 16 lanes hit 16 banks.
    __shared__ __align__(128) float lw[128 * 132];   // W transposed: lw[col][k]
    __shared__ __align__(128) float la[16 * 132];    // A tile: la[row][k]

    const int tid = threadIdx.x;
    const int rowbase = blockIdx.x * 16;

    // ---- TDM: DMA the 16x128 A tile into LDS (wave 0 issues; EXEC ignored).
    // D# per CDNA5 ISA ch.8: 2D tensor, data_size=4B, pad 4 DWORDs after
    // every 128 DWORDs (one row) -> 132-float LDS pitch. Rows beyond
    // tensor_dim1 read as zero (handles the final partial tile).
    if (tid < 32) {
        unsigned long long ga = (unsigned long long)(A + (size_t)rowbase * H);
        unsigned ldsa = (unsigned)(unsigned long long)la;  // low 32 bits = LDS offset
        v4u g0;
        g0.x = 1u;                                          // count=1 (valid), user mode
        g0.y = ldsa;                                        // lds_addr
        g0.z = (unsigned)ga;                                // global_addr[31:0]
        g0.w = ((unsigned)(ga >> 32) & 0x01FFFFFFu) | (2u << 30); // addr[56:32], type=2
        unsigned d1 = (unsigned)(M - rowbase);              // rows available from tile start
        v8i g1;
        g1[0] = (int)((2u << 16)      // data_size = 4 bytes
                    | (1u << 20)      // pad_enable
                    | (6u << 22)      // pad_interval: 128 DWORDs
                    | (3u << 25));    // pad_amount: 4 DWORDs
        g1[1] = (int)(((unsigned)H & 0xFFFFu) << 16);       // tensor_dim0 lo16 (=128)
        g1[2] = (int)(((unsigned)H >> 16) | ((d1 & 0xFFFFu) << 16)); // dim0 hi | dim1 lo
        g1[3] = (int)((d1 >> 16) | ((unsigned)H << 16));    // dim1 hi | tile_dim0 = 128
        g1[4] = 16;                                         // tile_dim1 = 16, tile_dim2 = 0
        g1[5] = H;                                          // tensor_dim0_stride lo32 = 128
        g1[6] = 0;                                          // dim0_stride hi | dim1_stride lo
        g1[7] = 0;                                          // dim1_stride hi
        v4i gz4 = {0, 0, 0, 0};
        v8i gz8 = {0, 0, 0, 0, 0, 0, 0, 0};
        // 6-arg toolchain form: (g0, g1, g2, g3, extra, cpol)
        __builtin_amdgcn_tensor_load_to_lds(g0, g1, gz4, gz4, gz8, 0);
        __builtin_amdgcn_s_wait_tensorcnt(0);
    }

    // ---- cooperative transposed stage of W into LDS (L2-hot, once per block)
    for (int i = tid; i < H * H; i += 256) {
        int r = i >> 7, c = i & 127;
        lw[c * 132 + r] = W[i];
    }
    __syncthreads();

    const int wave = tid >> 5;       // 0..7 -> N-tile
    const int lane = tid & 31;
    const int hsel = lane >> 4;      // 0: K pair {k,k+1}, 1: {k+2,k+3}
    const int rw   = lane & 15;
    const int colbase = wave * 16;
    const float* lap = la + rw * 132;
    const float* lwp = lw + (colbase + rw) * 132;

    v8f acc = {};
    #pragma unroll 8
    for (int k = 0; k < H; k += 4) {
        // ISA 7.12.2: 32-bit A 16x4 — lanes 0-15 hold K={k,k+1},
        // lanes 16-31 hold K={k+2,k+3}; M/N = lane&15. B mirrors A.
        int kk = k + 2 * hsel;
        v2f a, b;
        a.x = lap[kk];
        a.y = lap[kk + 1];
        b.x = lwp[kk];
        b.y = lwp[kk + 1];
        acc = __builtin_amdgcn_wmma_f32_16x16x4_f32(
            /*neg_a=*/false, a, /*neg_b=*/false, b,
            /*c_mod=*/(short)0, acc, /*reuse_a=*/false, /*reuse_b=*/false);
    }

    const int n = colbase + rw;
    const float bv = bias ? bias[n] : 0.0f;
    #pragma unroll
    for (int r = 0; r < 8; r++) {
        int m = r + 8 * hsel;            // D layout: VGPR r -> M = r + 8*half
        int rr = rowbase + m;
        if (rr < M) {
            float v = acc[r] + bv;
            if (act == 1) v = fmaxf(v, 0.0f);
            else if (act == 2) v = tanhf(v);
            out[(long)rr * H + n] = v;
        }
    }
}

// ---------------------------------------------------------------- GCN aggregation
// agg[n] = z[n] * dinv[n]^2 + conv_b    (self-loop term, also initializes agg)
__global__ void k_self_bias(const float* __restrict__ z, const float* __restrict__ dinv,
                            const float* __restrict__ bias, float* __restrict__ agg, int N) {
    int i = blockIdx.x * blockDim.x + threadIdx.x;   // over N*32 float4 groups
    if (i >= N * 32) return;
    int n  = i >> 5;
    int c4 = (i & 31) * 4;
    float d = dinv[n];
    float s = d * d;
    float4 zv = *(const float4*)(z + (long)n * H + c4);
    float4 bv = *(const float4*)(bias + c4);
    float4 o;
    o.x = zv.x * s + bv.x;
    o.y = zv.y * s + bv.y;
    o.z = zv.z * s + bv.z;
    o.w = zv.w * s + bv.w;
    *(float4*)(agg + (long)n * H + c4) = o;
}

// agg[dst] += z[src] * dinv[src]*dinv[dst], one wave32 per edge,
// 4 coalesced channels per lane -> global_atomic_add_f32.
__global__ __launch_bounds__(256)
void k_edge_agg(const int* __restrict__ src, const int* __restrict__ dst,
                const float* __restrict__ z, const float* __restrict__ dinv,
                float* __restrict__ agg, int E) {
    int gid = blockIdx.x * blockDim.x + threadIdx.x;
    int e = gid >> 5;
    int lane = gid & 31;
    if (e >= E) return;
    int s = src[e], d = dst[e];
    float nrm = dinv[s] * dinv[d];
    const float* zs = z + (long)s * H;
    float* ad = agg + (long)d * H;
    #pragma unroll
    for (int j = 0; j < 4; j++) {
        int c = lane + j * 32;
        atomicAdd(&ad[c], zs[c] * nrm);
    }
}

// ---------------------------------------------------------------- batchnorm
__global__ __launch_bounds__(256)
void k_bn_stats(const float* __restrict__ agg, float* __restrict__ gsum,
                float* __restrict__ gsq, int N) {
    __shared__ float ssum[256], ssq[256];
    int c  = threadIdx.x & 127;
    int rg = threadIdx.x >> 7;   // 0/1
    float s = 0.0f, q = 0.0f;
    for (int r = blockIdx.x * 2 + rg; r < N; r += gridDim.x * 2) {
        float v = agg[(long)r * H + c];
        s += v;
        q += v * v;
    }
    ssum[threadIdx.x] = s;
    ssq[threadIdx.x]  = q;
    __syncthreads();
    if (rg == 0) {
        atomicAdd(&gsum[c], ssum[c] + ssum[c + 128]);
        atomicAdd(&gsq[c],  ssq[c]  + ssq[c + 128]);
    }
}

__global__ void k_bn_final(const float* __restrict__ gsum, const float* __restrict__ gsq,
                           const float* __restrict__ g, const float* __restrict__ b,
                           float* __restrict__ scale, float* __restrict__ shift, int N) {
    int c = threadIdx.x;
    float invN = 1.0f / (float)N;
    float mean = gsum[c] * invN;
    float var  = gsq[c] * invN - mean * mean;
    float inv  = rsqrtf(var + EPS);
    float sc   = g[c] * inv;
    scale[c] = sc;
    shift[c] = b[c] - mean * sc;
}

// h = relu(agg*scale + shift) [+ residual]
__global__ void k_bn_apply(const float* __restrict__ agg, const float* __restrict__ scale,
                           const float* __restrict__ shift, const float* __restrict__ res,
                           float* __restrict__ h, int N) {
    int i = blockIdx.x * blockDim.x + threadIdx.x;
    if (i >= N * 32) return;
    int n  = i >> 5;
    int c4 = (i & 31) * 4;
    float4 a  = *(const float4*)(agg + (long)n * H + c4);
    float4 sc = *(const float4*)(scale + c4);
    float4 sh = *(const float4*)(shift + c4);
    float4 o;
    o.x = fmaxf(a.x * sc.x + sh.x, 0.0f);
    o.y = fmaxf(a.y * sc.y + sh.y, 0.0f);
    o.z = fmaxf(a.z * sc.z + sh.z, 0.0f);
    o.w = fmaxf(a.w * sc.w + sh.w, 0.0f);
    if (res) {
        float4 r = *(const float4*)(res + (long)n * H + c4);
        o.x += r.x; o.y += r.y; o.z += r.z; o.w += r.w;
    }
    *(float4*)(h + (long)n * H + c4) = o;
}

// ---------------------------------------------------------------- attention
__global__ void k_logits(const float* __restrict__ t, const float* __restrict__ w2,
                         const float* __restrict__ b2, float* __restrict__ l, int N) {
    int n = blockIdx.x * blockDim.x + threadIdx.x;
    if (n >= N) return;
    float s = b2[0];
    const float4* tp = (const float4*)(t + (long)n * H);
    const float4* wp = (const float4*)w2;
    #pragma unroll
    for (int j = 0; j < 32; j++) {
        float4 tv = tp[j], wv = wp[j];
        s += tv.x * wv.x + tv.y * wv.y + tv.z * wv.z + tv.w * wv.w;
    }
    l[n] = s;
}

__global__ __launch_bounds__(256)
void k_max(const float* __restrict__ l, float* __restrict__ gmax, int N) {
    __shared__ float sm[256];
    int tid = threadIdx.x;
    float m = -INFINITY;
    for (int i = blockIdx.x * 256 + tid; i < N; i += gridDim.x * 256)
        m = fmaxf(m, l[i]);
    sm[tid] = m;
    __syncthreads();
    for (int s = 128; s > 0; s >>= 1) {
        if (tid < s) sm[tid] = fmaxf(sm[tid], sm[tid + s]);
        __syncthreads();
    }
    if (tid == 0) atomicMaxF(gmax, sm[0]);
}

__global__ __launch_bounds__(256)
void k_expsum(const float* __restrict__ l, const float* __restrict__ gmax,
              float* __restrict__ e, float* __restrict__ gsum, int N) {
    __shared__ float sm[256];
    int tid = threadIdx.x;
    float mx = gmax[0];
    float s = 0.0f;
    for (int i = blockIdx.x * 256 + tid; i < N; i += gridDim.x * 256) {
        float v = __expf(l[i] - mx);
        e[i] = v;
        s += v;
    }
    sm[tid] = s;
    __syncthreads();
    for (int st = 128; st > 0; st >>= 1) {
        if (tid < st) sm[tid] += sm[tid + st];
        __syncthreads();
    }
    if (tid == 0) atomicAdd(gsum, sm[0]);
}

// ---------------------------------------------------------------- pooling
__global__ __launch_bounds__(256)
void k_pool(const float* __restrict__ h, const float* __restrict__ e,
            const float* __restrict__ gsum, const int* __restrict__ batch,
            float* __restrict__ psum, float* __restrict__ pcnt, int N) {
    int gid = blockIdx.x * blockDim.x + threadIdx.x;
    int n = gid >> 5, lane = gid & 31;
    if (n >= N) return;
    int b = batch[n];
    float w = e[n] / gsum[0];    // softmax weight
    const float* hp = h + (long)n * H;
    float* pp = psum + (long)b * H;
    #pragma unroll
    for (int j = 0; j < 4; j++) {
        int c = lane + j * 32;
        atomicAdd(&pp[c], hp[c] * w);
    }
    if (lane == 0) atomicAdd(&pcnt[b], 1.0f);
}

__global__ void k_pool_div(const float* __restrict__ psum, const float* __restrict__ pcnt,
                           float* __restrict__ pooled, int G) {
    int i = blockIdx.x * blockDim.x + threadIdx.x;
    if (i >= G * H) return;
    int g = i >> 7;
    pooled[i] = psum[i] / fmaxf(pcnt[g], 1.0f);
}

// ---------------------------------------------------------------- launch
static inline size_t alignup(size_t v) { return (v + 15) & ~(size_t)15; }

extern "C" void kernel_launch(void* const* d_in, const int* in_sizes, int n_in,
                              void* d_out, int out_size, void* d_ws, size_t ws_size,
                              hipStream_t stream) {
    const float* x      = (const float*)d_in[0];
    const int*   ei     = (const int*)d_in[1];     // [2, E] (src row, dst row)
    const int*   batch  = (const int*)d_in[2];
    const float* conv_w = (const float*)d_in[3];
    const float* conv_b = (const float*)d_in[4];
    const float* bn_g   = (const float*)d_in[5];
    const float* bn_b   = (const float*)d_in[6];
    const float* aw1    = (const float*)d_in[7];
    const float* ab1    = (const float*)d_in[8];
    const float* aw2    = (const float*)d_in[9];
    const float* ab2    = (const float*)d_in[10];
    const float* pw1    = (const float*)d_in[11];
    const float* pb1    = (const float*)d_in[12];
    const float* pw2    = (const float*)d_in[13];
    const float* pb2    = (const float*)d_in[14];

    const int N = in_sizes[0] / H;
    const int E = in_sizes[1] / 2;
    const int L = in_sizes[3] / (H * H);
    const int G = out_size / H;
    const int* src = ei;
    const int* dst = ei + E;
    float* out = (float*)d_out;

    // workspace carve-up (floats), 16-float aligned segments
    const size_t NH = (size_t)N * H;
    float* ws    = (float*)d_ws;
    size_t off = 0;
    float* h     = ws + off; off += alignup(NH);
    float* z     = ws + off; off += alignup(NH);     // also reused as attention t
    float* agg   = ws + off; off += alignup(NH);
    float* dinv  = ws + off; off += alignup(N);
    float* lg    = ws + off; off += alignup(N);
    float* ee    = ws + off; off += alignup(N);
    float* gsum  = ws + off; off += alignup(H);
    float* gsq   = ws + off; off += alignup(H);
    float* scale = ws + off; off += alignup(H);
    float* shift = ws + off; off += alignup(H);
    float* gmax  = ws + off; off += alignup(1);
    float* gse   = ws + off; off += alignup(1);
    float* psum  = ws + off; off += alignup((size_t)G * H);
    float* pcnt  = ws + off; off += alignup(G);
    float* pooled= ws + off; off += alignup((size_t)G * H);
    float* mlp1  = ws + off; off += alignup((size_t)G * H);
    (void)ws_size; (void)n_in;

    const int B256 = 256;
    const int gN    = (N + B256 - 1) / B256;
    const int gN32  = ((size_t)N * 32 + B256 - 1) / B256;
    const int gE    = (E + B256 - 1) / B256;
    const int gE32  = (E + 7) / 8;                 // 8 edges (waves) per block
    const int gTile = (N + 15) / 16;
    const int gGT   = (G + 15) / 16;

    // ---- GCN normalization: deg (with self-loop) -> d^-1/2
    k_fill<<<gN, B256, 0, stream>>>(dinv, 1.0f, N);
    k_deg_count<<<gE, B256, 0, stream>>>(dst, dinv, E);
    k_dinv<<<gN, B256, 0, stream>>>(dinv, N);

    // ---- GCN layers
    const float* hin = x;
    for (int i = 0; i < L; i++) {
        k_gemm128<<<gTile, B256, 0, stream>>>(hin, conv_w + (size_t)i * H * H,
                                              nullptr, z, N, 0);
        k_self_bias<<<gN32, B256, 0, stream>>>(z, dinv, conv_b + i * H, agg, N);
        k_edge_agg<<<gE32, B256, 0, stream>>>(src, dst, z, dinv, agg, E);
        k_fill<<<1, 2 * H, 0, stream>>>(gsum, 0.0f, 2 * H);  // gsum & gsq contiguous
        k_bn_stats<<<512, B256, 0, stream>>>(agg, gsum, gsq, N);
        k_bn_final<<<1, H, 0, stream>>>(gsum, gsq, bn_g + i * H, bn_b + i * H,
                                        scale, shift, N);
        k_bn_apply<<<gN32, B256, 0, stream>>>(agg, scale, shift,
                                              (i > 0) ? h : nullptr, h, N);
        hin = h;
    }

    // ---- attention: t = tanh(h@w1+b1); logits = t@w2+b2; softmax over N
    k_gemm128<<<gTile, B256, 0, stream>>>(h, aw1, ab1, z, N, 2);
    k_logits<<<gN, B256, 0, stream>>>(z, aw2, ab2, lg, N);
    k_fill<<<1, 32, 0, stream>>>(gmax, -INFINITY, 1);
    k_fill<<<1, 32, 0, stream>>>(gse, 0.0f, 1);
    k_max<<<512, B256, 0, stream>>>(lg, gmax, N);
    k_expsum<<<512, B256, 0, stream>>>(lg, gmax, ee, gse, N);

    // ---- weighted global mean pool
    k_fill<<<(G * H + B256 - 1) / B256, B256, 0, stream>>>(psum, 0.0f, G * H);
    k_fill<<<1, G, 0, stream>>>(pcnt, 0.0f, G);
    k_pool<<<(N * 32 + B256 - 1) / B256, B256, 0, stream>>>(h, ee, gse, batch,
                                                            psum, pcnt, N);
    k_pool_div<<<(G * H + B256 - 1) / B256, B256, 0, stream>>>(psum, pcnt, pooled, G);

    // ---- projection MLP (WMMA again)
    k_gemm128<<<gGT, B256, 0, stream>>>(pooled, pw1, pb1, mlp1, G, 1);
    k_gemm128<<<gGT, B256, 0, stream>>>(mlp1, pw2, pb2, out, G, 0);
}